// Model_39676907881133
// MI455X (gfx1250) — compile-verified
//
#include <hip/hip_runtime.h>

typedef __attribute__((ext_vector_type(2))) float v2f;
typedef __attribute__((ext_vector_type(8))) float v8f;

__device__ __forceinline__ v8f wmma_f32(v2f a, v2f b, v8f c) {
    // (neg_a, A, neg_b, B, c_mod, C, reuse_a, reuse_b)
    return __builtin_amdgcn_wmma_f32_16x16x4_f32(false, a, false, b,
                                                 (short)0, c, false, false);
}

// Unconditional A-fragment read from a zero-padded 4x4 tile (row stride 4).
// lane L: M = L&15 (rows >=3 read the all-zero row 3), K = 2*(L>>4)+{0,1}.
__device__ __forceinline__ v2f a_frag_pad(const float* pad, int mc, int k) {
    v2f a;
    a.x = pad[mc * 4 + k];
    a.y = pad[mc * 4 + k + 1];
    return a;
}

// Unconditional B-fragment read: B(k, n) = pad[k*4 + min(n,3)] (col 3 is zero).
__device__ __forceinline__ v2f b_frag_pad(const float* pad, int nc, int k) {
    v2f b;
    b.x = pad[k * 4 + nc];
    b.y = pad[(k + 1) * 4 + nc];
    return b;
}

__global__ __launch_bounds__(32)
void model_3x3_chain_kernel(const float* __restrict__ in1,
                            const float* __restrict__ in2,
                            const float* __restrict__ in3,
                            const float* __restrict__ in4,
                            const float* __restrict__ in5,
                            const float* __restrict__ in6,
                            float* __restrict__ out) {
    // 6 input tiles (4x4, zero-padded) + s tile + t18 tile = 128 floats.
    __shared__ float pads[128];
    float* const pin = pads;          // pin + 16*j : input j
    float* const sp  = pads + 96;     // s   = in1@in3 + in2@in4 + in5@in6
    float* const tp  = pads + 112;    // t18 = in1@in6 + in2@in5 + in3@in6

    const int lane = threadIdx.x;

    // Zero-init all padding (4 stores/lane, no divergence).
    pads[lane]      = 0.0f;
    pads[lane + 32] = 0.0f;
    pads[lane + 64] = 0.0f;
    pads[lane + 96] = 0.0f;
    __syncthreads();

    // Scatter each 3x3 into its padded 4x4 tile.
    if (lane < 9) {
        const int r = lane / 3;
        const int c = lane - 3 * r;
        const int o = r * 4 + c;
        pin[0 * 16 + o] = in1[lane];
        pin[1 * 16 + o] = in2[lane];
        pin[2 * 16 + o] = in3[lane];
        pin[3 * 16 + o] = in4[lane];
        pin[4 * 16 + o] = in5[lane];
        pin[5 * 16 + o] = in6[lane];
    }
    __syncthreads();

    // Per-lane fragment coordinates (branch-free).
    const int m  = lane & 15;
    const int mc = (m < 3) ? m : 3;          // clamped row/col -> zero padding
    const int k  = (lane >> 4) << 1;         // K base: 0 or 2

    const v2f A1 = a_frag_pad(pin + 0 * 16, mc, k);
    const v2f A2 = a_frag_pad(pin + 1 * 16, mc, k);
    const v2f A3 = a_frag_pad(pin + 2 * 16, mc, k);
    const v2f A5 = a_frag_pad(pin + 4 * 16, mc, k);
    const v2f B3 = b_frag_pad(pin + 2 * 16, mc, k);
    const v2f B4 = b_frag_pad(pin + 3 * 16, mc, k);
    const v2f B5 = b_frag_pad(pin + 4 * 16, mc, k);
    const v2f B6 = b_frag_pad(pin + 5 * 16, mc, k);

    // s = in1@in3 + in2@in4 + in5@in6
    v8f s_acc = {};
    s_acc = wmma_f32(A1, B3, s_acc);
    s_acc = wmma_f32(A2, B4, s_acc);
    s_acc = wmma_f32(A5, B6, s_acc);

    // t18 = in1@in6 + in2@in5 + in3@in6
    v8f t_acc = {};
    t_acc = wmma_f32(A1, B6, t_acc);
    t_acc = wmma_f32(A2, B5, t_acc);
    t_acc = wmma_f32(A3, B6, t_acc);

    // C/D layout: VGPR r, lanes 0..15 hold (M=r, N=lane). Live 3x3 block is
    // acc[0..2] on lanes 0..2. Spill into zero-padded 4x4 tiles.
    if (lane < 3) {
        sp[0 * 4 + lane] = s_acc[0];
        sp[1 * 4 + lane] = s_acc[1];
        sp[2 * 4 + lane] = s_acc[2];
        tp[0 * 4 + lane] = t_acc[0];
        tp[1 * 4 + lane] = t_acc[1];
        tp[2 * 4 + lane] = t_acc[2];
    }
    __syncthreads();

    // Final: (2s) @ (t18 + s), fragments built unconditionally.
    v2f At, Bt;
    At.x = 2.0f * sp[mc * 4 + k];
    At.y = 2.0f * sp[mc * 4 + k + 1];
    Bt.x = tp[k * 4 + mc]       + sp[k * 4 + mc];
    Bt.y = tp[(k + 1) * 4 + mc] + sp[(k + 1) * 4 + mc];

    v8f d = {};
    d = wmma_f32(At, Bt, d);

    if (lane < 3) {
        out[0 * 3 + lane] = d[0];
        out[1 * 3 + lane] = d[1];
        out[2 * 3 + lane] = d[2];
    }
}

extern "C" void kernel_launch(void* const* d_in, const int* in_sizes, int n_in,
                              void* d_out, int out_size, void* d_ws, size_t ws_size,
                              hipStream_t stream) {
    (void)in_sizes; (void)n_in; (void)out_size; (void)d_ws; (void)ws_size;
    model_3x3_chain_kernel<<<1, 32, 0, stream>>>(
        (const float*)d_in[0], (const float*)d_in[1], (const float*)d_in[2],
        (const float*)d_in[3], (const float*)d_in[4], (const float*)d_in[5],
        (float*)d_out);
}